// Net_5471788335191
// MI455X (gfx1250) — compile-verified
//
#include <hip/hip_runtime.h>
#include <stdint.h>

typedef __attribute__((ext_vector_type(2))) float v2f;
typedef __attribute__((ext_vector_type(8))) float v8f;

#define N_NODES 100000
#define N_EDGES 3200000
#define F_INDIM 300
#define HID     16
#define NCLS    10

static __device__ __forceinline__ void atomic_add_f32(float* p, float v) {
    __hip_atomic_fetch_add(p, v, __ATOMIC_RELAXED, __HIP_MEMORY_SCOPE_AGENT);
}

// ---------- degree / norm ----------
__global__ void k_init_deg(float* deg) {
    int i = blockIdx.x * blockDim.x + threadIdx.x;
    if (i < N_NODES) deg[i] = 1.0f;   // self-loop weight
}

__global__ void k_accum_deg(const int* __restrict__ ei, const float* __restrict__ w,
                            float* __restrict__ deg) {
    int e = blockIdx.x * blockDim.x + threadIdx.x;
    if (e < N_EDGES) atomic_add_f32(&deg[ei[N_EDGES + e]], w[e]);  // col = targets
}

__global__ void k_dinv(float* deg) {
    int i = blockIdx.x * blockDim.x + threadIdx.x;
    if (i < N_NODES) {
        float d = deg[i];
        deg[i] = (d > 0.0f) ? rsqrtf(d) : 0.0f;   // in place: deg -> dinv
    }
}

// ---------- pack W1 for b64 B-operand loads ----------
// P[j][m] = { W1[2j][m], W1[2j+1][m] },  j in [0,150), m in [0,16)
__global__ void k_packW1(const float* __restrict__ W1, v2f* __restrict__ P) {
    int t = blockIdx.x * blockDim.x + threadIdx.x;
    if (t < (F_INDIM / 2) * HID) {
        int j = t >> 4;
        int m = t & 15;
        v2f p;
        p.x = W1[(2 * j) * HID + m];
        p.y = W1[(2 * j + 1) * HID + m];
        P[t] = p;
    }
}

// ---------- GEMM1: h0 = x @ W1 via V_WMMA_F32_16X16X4_F32 ----------
// one wave handles TWO 16-row tiles (shared B fragment); K=300 in steps of 4.
__global__ __launch_bounds__(256) void k_gemm1(const float* __restrict__ x,
                                               const v2f* __restrict__ P,
                                               float* __restrict__ h0) {
    const int lane = threadIdx.x & 31;
    const int wave = threadIdx.x >> 5;
    const int job  = blockIdx.x * 8 + wave;          // pair-of-tiles index
    const int tile0 = job * 2;
    if (tile0 >= N_NODES / 16) return;

    const int m    = lane & 15;          // row within tile / column of B
    const int hsel = lane >> 4;          // 0: K+0,1  1: K+2,3
    const int koff = hsel << 1;

    const float* xr0 = x + (size_t)(tile0 * 16 + m) * F_INDIM + koff;
    const float* xr1 = xr0 + (size_t)16 * F_INDIM;
    const v2f*   bp  = P + (size_t)hsel * HID + m;   // P[(k>>1)+hsel][m]

    v8f c0 = {};
    v8f c1 = {};
    #pragma unroll 5
    for (int k = 0; k < F_INDIM; k += 4) {
        v2f a0; a0.x = xr0[0]; a0.y = xr0[1];
        v2f a1; a1.x = xr1[0]; a1.y = xr1[1];
        v2f b = *bp;                                  // single b64 load, shared
        c0 = __builtin_amdgcn_wmma_f32_16x16x4_f32(
                 false, a0, false, b, (short)0, c0, false, false);
        c1 = __builtin_amdgcn_wmma_f32_16x16x4_f32(
                 false, a1, false, b, (short)0, c1, false, false);
        xr0 += 4;
        xr1 += 4;
        bp  += 2 * HID;                               // advance 2 pair-rows
    }

    // D layout: VGPR v -> row M=v (lanes 0-15) / M=v+8 (lanes 16-31), N=lane&15
    const int hi8 = hsel << 3;
    float* out0 = h0 + (size_t)(tile0 * 16 + hi8) * HID + m;
    float* out1 = out0 + (size_t)16 * HID;
    #pragma unroll
    for (int v = 0; v < 8; ++v) out0[v * HID] = c0[v];
    #pragma unroll
    for (int v = 0; v < 8; ++v) out1[v * HID] = c1[v];
}

// ---------- conv1: self-loop init + edge scatter (16 feats) ----------
__global__ void k_selfloop16(const float* __restrict__ h0, const float* __restrict__ dinv,
                             float* __restrict__ h1) {
    unsigned t = blockIdx.x * blockDim.x + threadIdx.x;
    if (t < (unsigned)N_NODES * 16u) {
        int i = t >> 4;
        float di = dinv[i];
        h1[t] = di * di * h0[t];
    }
}

__global__ void k_scatter16(const int* __restrict__ ei, const float* __restrict__ w,
                            const float* __restrict__ dinv,
                            const float* __restrict__ h0, float* __restrict__ h1) {
    unsigned t = blockIdx.x * blockDim.x + threadIdx.x;
    if (t >= (unsigned)N_EDGES * 16u) return;
    int e = (int)(t >> 4);
    int f = (int)(t & 15u);
    int r = ei[e];
    int cdst = ei[N_EDGES + e];
    float nrm = dinv[r] * w[e] * dinv[cdst];
    atomic_add_f32(&h1[(size_t)cdst * 16 + f], h0[(size_t)r * 16 + f] * nrm);
}

// ---------- layer2: h2 = relu(h1 + b1) @ W2 ----------
__global__ __launch_bounds__(256) void k_layer2(const float* __restrict__ h1,
                                                const float* __restrict__ b1,
                                                const float* __restrict__ W2,
                                                float* __restrict__ h2) {
    __shared__ float sW[HID * NCLS];
    __shared__ float sb[HID];
    if (threadIdx.x < HID * NCLS) sW[threadIdx.x] = W2[threadIdx.x];
    if (threadIdx.x < HID)        sb[threadIdx.x] = b1[threadIdx.x];
    __syncthreads();

    int i = blockIdx.x * blockDim.x + threadIdx.x;
    if (i >= N_NODES) return;

    float hv[HID];
    #pragma unroll
    for (int f = 0; f < HID; ++f)
        hv[f] = fmaxf(h1[(size_t)i * HID + f] + sb[f], 0.0f);

    #pragma unroll
    for (int cc = 0; cc < NCLS; ++cc) {
        float s = 0.0f;
        #pragma unroll
        for (int f = 0; f < HID; ++f) s = fmaf(hv[f], sW[f * NCLS + cc], s);
        h2[(size_t)i * NCLS + cc] = s;
    }
}

// ---------- conv2: self-loop init (into d_out) + edge scatter (10 feats) ----------
__global__ void k_selfloop10(const float* __restrict__ h2, const float* __restrict__ dinv,
                             float* __restrict__ out) {
    unsigned t = blockIdx.x * blockDim.x + threadIdx.x;
    if (t < (unsigned)N_NODES * (unsigned)NCLS) {
        int i = t / NCLS;
        float di = dinv[i];
        out[t] = di * di * h2[t];
    }
}

__global__ void k_scatter10(const int* __restrict__ ei, const float* __restrict__ w,
                            const float* __restrict__ dinv,
                            const float* __restrict__ h2, float* __restrict__ out) {
    unsigned t = blockIdx.x * blockDim.x + threadIdx.x;
    if (t >= (unsigned)N_EDGES * (unsigned)NCLS) return;
    int e = (int)(t / NCLS);
    int cc = (int)(t % NCLS);
    int r = ei[e];
    int cdst = ei[N_EDGES + e];
    float nrm = dinv[r] * w[e] * dinv[cdst];
    atomic_add_f32(&out[(size_t)cdst * NCLS + cc], h2[(size_t)r * NCLS + cc] * nrm);
}

// ---------- finalize: out = log_softmax(out + b2) ----------
__global__ void k_finalize(float* __restrict__ out, const float* __restrict__ b2) {
    int i = blockIdx.x * blockDim.x + threadIdx.x;
    if (i >= N_NODES) return;
    float v[NCLS];
    float mx = -3.4e38f;
    #pragma unroll
    for (int cc = 0; cc < NCLS; ++cc) {
        v[cc] = out[(size_t)i * NCLS + cc] + b2[cc];
        mx = fmaxf(mx, v[cc]);
    }
    float s = 0.0f;
    #pragma unroll
    for (int cc = 0; cc < NCLS; ++cc) s += expf(v[cc] - mx);
    float ls = mx + logf(s);
    #pragma unroll
    for (int cc = 0; cc < NCLS; ++cc) out[(size_t)i * NCLS + cc] = v[cc] - ls;
}

extern "C" void kernel_launch(void* const* d_in, const int* in_sizes, int n_in,
                              void* d_out, int out_size, void* d_ws, size_t ws_size,
                              hipStream_t stream) {
    const float* x  = (const float*)d_in[0];
    const int*   ei = (const int*)d_in[1];     // [2][E] row-major: row, col
    const float* ew = (const float*)d_in[2];
    const float* W1 = (const float*)d_in[3];
    const float* b1 = (const float*)d_in[4];
    const float* W2 = (const float*)d_in[5];
    const float* b2 = (const float*)d_in[6];
    float* out = (float*)d_out;

    float* ws  = (float*)d_ws;
    float* deg = ws;                       // N floats, becomes dinv in place
    float* h0  = ws + 100352;              // N*16
    float* h1  = h0 + (size_t)N_NODES * HID;
    float* h2  = h1 + (size_t)N_NODES * HID;
    v2f*   P   = (v2f*)(h2 + (size_t)N_NODES * NCLS);   // 150*16 float2 (19.2 KB)

    const int B = 256;

    k_init_deg<<<(N_NODES + B - 1) / B, B, 0, stream>>>(deg);
    k_accum_deg<<<(N_EDGES + B - 1) / B, B, 0, stream>>>(ei, ew, deg);
    k_dinv<<<(N_NODES + B - 1) / B, B, 0, stream>>>(deg);

    k_packW1<<<((F_INDIM / 2) * HID + B - 1) / B, B, 0, stream>>>(W1, P);

    // GEMM1: 6250 tiles, 2 tiles per wave -> 3125 jobs, 8 waves per block
    k_gemm1<<<(N_NODES / 32 + 7) / 8, B, 0, stream>>>(x, P, h0);

    k_selfloop16<<<((unsigned)N_NODES * 16u + B - 1) / B, B, 0, stream>>>(h0, deg, h1);
    k_scatter16<<<((unsigned)N_EDGES * 16u + B - 1) / B, B, 0, stream>>>(ei, ew, deg, h0, h1);

    k_layer2<<<(N_NODES + B - 1) / B, B, 0, stream>>>(h1, b1, W2, h2);

    k_selfloop10<<<((unsigned)N_NODES * NCLS + B - 1) / B, B, 0, stream>>>(h2, deg, out);
    k_scatter10<<<((unsigned)N_EDGES * NCLS + B - 1) / B, B, 0, stream>>>(ei, ew, deg, h2, out);

    k_finalize<<<(N_NODES + B - 1) / B, B, 0, stream>>>(out, b2);
}